// FraudSNNPyTorch_90915867722005
// MI455X (gfx1250) — compile-verified
//
#include <hip/hip_runtime.h>

// MI455X / gfx1250 fused SNN forward.
// Strategy: memory-bound problem (636MB of mandatory rec writes vs ~16 GFLOP f32);
// use exact V_WMMA_F32_16X16X4_F32 for both GEMMs, keep membranes in VGPRs, cur1/spike
// staging entirely in LDS (320KB/WGP), stream only the recordings to HBM (NT).

typedef __attribute__((ext_vector_type(2))) float v2f;
typedef __attribute__((ext_vector_type(4))) float f4;
typedef __attribute__((ext_vector_type(8))) float v8f;

#define BATCH 32768
#define DIN   256
#define H1    128
#define H2    64

// LDS row strides (floats), chosen for 64-bank conflict freedom:
//  - stride 132 (mod 64 == 4): D-layout element access (lanes16-31 offset by
//    8*132 % 64 = 32 banks) AND b64 A-operand reads (banks 4*row + k) both clean.
//  - stride 68 (mod 64 == 4): same property for the staging / s2 arrays.
#define XS_ST 68
#define W1_ST 68
#define C1_ST 132
#define S1_ST 132
#define W2_ST 132
#define S2_ST 68

__global__ __launch_bounds__(256) void snn_fused_kernel(
    const float* __restrict__ x,  const float* __restrict__ W1, const float* __restrict__ b1,
    const float* __restrict__ W2, const float* __restrict__ b2,
    const float* __restrict__ W3, const float* __restrict__ b3,
    const int* __restrict__ nsteps, float* __restrict__ dout)
{
    __shared__ __align__(16) float xs[128 * XS_ST];     // x K-chunk stage     (~34KB)
    __shared__ __align__(16) float w1s[128 * W1_ST];    // W1 K-chunk stage    (~34KB)
    __shared__ __align__(16) float cur1s[128 * C1_ST];  // layer-1 currents    (~66KB)
    __shared__ __align__(16) float s1s[128 * S1_ST];    // layer-1 spikes      (~66KB)
    __shared__ __align__(16) float w2s[64 * W2_ST];     // W2 resident        (~33KB)
    __shared__ __align__(16) float s2s[128 * S2_ST];    // layer-2 spikes      (~34KB)
    __shared__ __align__(16) float w3s[2 * 64];

    const int tid = threadIdx.x;
    const int wv  = tid >> 5;        // wave id: owns rows [16*wv, 16*wv+16)
    const int ln  = tid & 31;
    const int lm  = ln & 15;         // N / M index within half-wave
    const int hi  = ln >> 4;         // half-wave select
    const int rowBase = blockIdx.x * 128;

    // ---- Resident weights: W2 -> LDS (64x128), W3 -> LDS (2x64) ----
    {
        int r = tid >> 2, q = tid & 3;
        const float* src = W2 + r * H1 + q * 32;
        float* dst = w2s + r * W2_ST + q * 32;
        #pragma unroll
        for (int i = 0; i < 8; ++i)
            *(f4*)(dst + 4 * i) = *(const f4*)(src + 4 * i);
        if (tid < 128) w3s[tid] = W3[tid];
    }

    // ---- Phase 1: cur1 = x @ W1^T + b1 for this block's 128 rows ----
    v8f acc[8];
    #pragma unroll
    for (int g = 0; g < 8; ++g) {
        float bv = b1[16 * g + lm];
        #pragma unroll
        for (int r = 0; r < 8; ++r) acc[g][r] = bv;
    }

    for (int kc = 0; kc < 4; ++kc) {            // K chunks of 64 over DIN=256
        __syncthreads();
        {   // stage x[128 x 64] and W1[128 x 64] chunks
            int r = tid >> 1, h = tid & 1;
            const float* xsrc = x  + (size_t)(rowBase + r) * DIN + kc * 64 + h * 32;
            const float* wsrc = W1 + (size_t)r * DIN            + kc * 64 + h * 32;
            float* xdst = xs  + r * XS_ST + h * 32;
            float* wdst = w1s + r * W1_ST + h * 32;
            #pragma unroll
            for (int i = 0; i < 8; ++i) {
                *(f4*)(xdst + 4 * i) = *(const f4*)(xsrc + 4 * i);
                *(f4*)(wdst + 4 * i) = *(const f4*)(wsrc + 4 * i);
            }
        }
        __syncthreads();
        #pragma unroll
        for (int k = 0; k < 16; ++k) {          // K steps of 4
            // A (16x4 f32): lane = M, vgpr pair = K; per-lane K pair at 4k+2*hi
            v2f a = *(const v2f*)&xs[(wv * 16 + lm) * XS_ST + 4 * k + 2 * hi];
            #pragma unroll
            for (int g = 0; g < 8; ++g) {       // 8 N-tiles -> H1=128
                v2f b = *(const v2f*)&w1s[(16 * g + lm) * W1_ST + 4 * k + 2 * hi];
                acc[g] = __builtin_amdgcn_wmma_f32_16x16x4_f32(
                    false, a, false, b, (short)0, acc[g], false, false);
            }
        }
    }
    // Park cur1 in LDS (own rows only; no cross-wave readers -> no barrier needed
    // beyond wave-internal DS ordering).
    #pragma unroll
    for (int g = 0; g < 8; ++g)
        #pragma unroll
        for (int r = 0; r < 8; ++r)
            cur1s[(wv * 16 + r + 8 * hi) * C1_ST + 16 * g + lm] = acc[g][r];
    __syncthreads();   // covers w2s/w3s visibility for all waves

    // ---- Phase 2: time loop (barrier-free; each wave independent) ----
    const int T = *nsteps;
    float* outp = dout;
    float* rec1 = dout + (size_t)BATCH * 2;
    float* rec2 = rec1 + (size_t)T * BATCH * H1;
    float* rec3 = rec2 + (size_t)T * BATCH * H2;

    v8f m1[8], m2[4];
    #pragma unroll
    for (int g = 0; g < 8; ++g)
        #pragma unroll
        for (int r = 0; r < 8; ++r) m1[g][r] = 0.0f;
    #pragma unroll
    for (int n = 0; n < 4; ++n)
        #pragma unroll
        for (int r = 0; r < 8; ++r) m2[n][r] = 0.0f;

    float b2v[4];
    #pragma unroll
    for (int n = 0; n < 4; ++n) b2v[n] = b2[16 * n + lm];

    const int r3 = ln >> 1, c3 = ln & 1;       // layer-3 lane -> (row, out col)
    const float b3v = b3[c3];
    float m3 = 0.0f, out3 = 0.0f;

    for (int t = 0; t < T; ++t) {
        // Layer 1 LIF (elementwise in WMMA C/D layout) + record + stage spikes
        #pragma unroll
        for (int g = 0; g < 8; ++g) {
            #pragma unroll
            for (int r = 0; r < 8; ++r) {
                int row = wv * 16 + r + 8 * hi;
                float cur = cur1s[row * C1_ST + 16 * g + lm];
                float m = m1[g][r];
                float reset = (m > 1.0f) ? 1.0f : 0.0f;   // spike(old_mem - 1)
                m = 0.9f * m + cur - reset;
                float s = (m > 1.0f) ? 1.0f : 0.0f;
                m1[g][r] = m;
                s1s[row * S1_ST + 16 * g + lm] = s;
                __builtin_nontemporal_store(
                    s, &rec1[((size_t)t * BATCH + rowBase + row) * H1 + 16 * g + lm]);
            }
        }

        // Layer 2: cur2 = s1 @ W2^T + b2 via f32 WMMA (K=128 -> 32 steps of 4)
        v8f a2[4];
        #pragma unroll
        for (int n = 0; n < 4; ++n)
            #pragma unroll
            for (int r = 0; r < 8; ++r) a2[n][r] = b2v[n];
        #pragma unroll 4
        for (int k = 0; k < 32; ++k) {
            v2f a = *(const v2f*)&s1s[(wv * 16 + lm) * S1_ST + 4 * k + 2 * hi];
            #pragma unroll
            for (int n = 0; n < 4; ++n) {
                v2f b = *(const v2f*)&w2s[(16 * n + lm) * W2_ST + 4 * k + 2 * hi];
                a2[n] = __builtin_amdgcn_wmma_f32_16x16x4_f32(
                    false, a, false, b, (short)0, a2[n], false, false);
            }
        }

        // Layer 2 LIF + record + stage spikes
        #pragma unroll
        for (int n = 0; n < 4; ++n) {
            #pragma unroll
            for (int r = 0; r < 8; ++r) {
                int row = wv * 16 + r + 8 * hi;
                float cur = a2[n][r];
                float m = m2[n][r];
                float reset = (m > 1.0f) ? 1.0f : 0.0f;
                m = 0.9f * m + cur - reset;
                float s = (m > 1.0f) ? 1.0f : 0.0f;
                m2[n][r] = m;
                s2s[row * S2_ST + 16 * n + lm] = s;
                __builtin_nontemporal_store(
                    s, &rec2[((size_t)t * BATCH + rowBase + row) * H2 + 16 * n + lm]);
            }
        }

        // Layer 3: N=2 too small for WMMA; per-lane 64-MAC dot (LDS broadcasts)
        float cur3 = b3v;
        #pragma unroll
        for (int j = 0; j < 64; j += 4) {
            f4 sv = *(const f4*)&s2s[(wv * 16 + r3) * S2_ST + j];
            f4 wvv = *(const f4*)&w3s[c3 * 64 + j];
            cur3 += sv.x * wvv.x + sv.y * wvv.y + sv.z * wvv.z + sv.w * wvv.w;
        }
        {
            float m = m3;
            float reset = (m > 1.0f) ? 1.0f : 0.0f;
            m = 0.9f * m + cur3 - reset;
            float s = (m > 1.0f) ? 1.0f : 0.0f;
            m3 = m;
            out3 += s;
            __builtin_nontemporal_store(
                s, &rec3[(size_t)t * (BATCH * 2) + (rowBase + wv * 16 + r3) * 2 + c3]);
        }
    }

    // output = rec3.sum(axis=0)
    outp[(rowBase + wv * 16 + r3) * 2 + c3] = out3;
}

extern "C" void kernel_launch(void* const* d_in, const int* in_sizes, int n_in,
                              void* d_out, int out_size, void* d_ws, size_t ws_size,
                              hipStream_t stream) {
    const float* x  = (const float*)d_in[0];
    const float* W1 = (const float*)d_in[1];
    const float* b1 = (const float*)d_in[2];
    const float* W2 = (const float*)d_in[3];
    const float* b2 = (const float*)d_in[4];
    const float* W3 = (const float*)d_in[5];
    const float* b3 = (const float*)d_in[6];
    const int* nsteps = (const int*)d_in[7];
    float* out = (float*)d_out;

    // 32768 rows / 128 rows per block = 256 blocks; 256 threads = 8 wave32.
    snn_fused_kernel<<<dim3(BATCH / 128), dim3(256), 0, stream>>>(
        x, W1, b1, W2, b2, W3, b3, nsteps, out);
}